// RadiusGraph_82480551952813
// MI455X (gfx1250) — compile-verified
//
#include <hip/hip_runtime.h>
#include <stdint.h>

typedef __attribute__((ext_vector_type(2))) float v2f;
typedef __attribute__((ext_vector_type(8))) float v8f;

#define CAP_MAX 64

// ---------------------------------------------------------------------------
// Phase 1: 16x16 distance tiles via V_WMMA_F32_16X16X4_F32.
//   d2(m,n) = |q_m|^2 + |r_n|^2 - 2 q_m . r_n
//   A (16x4 f32): rows = queries, K = [qx,qy,qz,0]
//       lane layout: lanes 0-15 M=0-15; VGPR0 holds K=0 (lo half)/K=2 (hi half),
//       VGPR1 holds K=1 / K=3.
//   B (4x16 f32): cols = refs, K = -2*[rx,ry,rz,0], mirrored layout.
//   C/D (16x16 f32): VGPR v, lane l -> M = v + 8*(l>=16), N = l%16.
//   C is pre-seeded with |q|^2 + |r|^2 so D is d2 directly.
// Valid (same-batch, d2<=r2) pairs are appended into per-query buckets.
// ---------------------------------------------------------------------------
__global__ __launch_bounds__(256)
void rg_phase1(const float* __restrict__ ref, const float* __restrict__ query,
               const int* __restrict__ radius, const int* __restrict__ sortflag,
               int NQ, int NR, int CAP,
               int* __restrict__ count, unsigned long long* __restrict__ cand)
{
    __shared__ float qx[16], qy[16], qz[16], qb[16], qn[16];

    const int tid  = threadIdx.x;
    const int lane = tid & 31;
    const int w    = tid >> 5;          // wave id 0..7 (wave32)
    const int q0   = blockIdx.x * 16;   // query tile base

    if (tid < 16) {
        int qg = q0 + tid;
        float4 qv = (qg < NQ) ? ((const float4*)query)[qg]
                              : make_float4(-1.0f, 1.0e6f, 1.0e6f, 1.0e6f);
        qb[tid] = qv.x; qx[tid] = qv.y; qy[tid] = qv.z; qz[tid] = qv.w;
        qn[tid] = qv.y*qv.y + qv.z*qv.z + qv.w*qv.w;
    }
    __syncthreads();

    const float rad = (float)radius[0];
    const float r2  = rad * rad;
    const int   sbd = sortflag[0];

    const int m  = lane & 15;   // column within tile (N) and query row for A
    const int hi = lane >> 4;   // lane half selects K pair / M offset

    // A operand: constant across the ref sweep.
    v2f a;
    a.x = hi ? qz[m] : qx[m];   // K = 2 : 0
    a.y = hi ? 0.0f  : qy[m];   // K = 3 : 1

    // Hoist per-row (M) query data for the C seed / validity test.
    float qnr[8], qbr[8];
    int   qgv[8];
#pragma unroll
    for (int v = 0; v < 8; ++v) {
        int mq = v + 8 * hi;
        qnr[v] = qn[mq];
        qbr[v] = qb[mq];
        qgv[v] = q0 + mq;
    }

    for (int r0 = w * 16; r0 < NR; r0 += 128) {
        if (r0 + 128 < NR)   // warm L2/L0 for the next tile of this wave
            __builtin_prefetch(&((const float4*)ref)[r0 + 128 + m], 0, 3);

        // Every lane loads ref (r0 + lane%16): lane-halves duplicate -> L0 hits.
        int rg = r0 + m;
        float4 rv = (rg < NR) ? ((const float4*)ref)[rg]
                              : make_float4(-2.0f, 1.0e6f, 1.0e6f, 1.0e6f);
        const float rb_ = rv.x;
        const float rn_ = rv.y*rv.y + rv.z*rv.z + rv.w*rv.w;

        v2f b;
        b.x = hi ? (-2.0f * rv.w) : (-2.0f * rv.y);  // K = 2 : 0
        b.y = hi ? 0.0f           : (-2.0f * rv.z);  // K = 3 : 1

        v8f c;
#pragma unroll
        for (int v = 0; v < 8; ++v) c[v] = qnr[v] + rn_;

        // Wave-uniform control flow here -> EXEC all ones as WMMA requires.
        v8f d = __builtin_amdgcn_wmma_f32_16x16x4_f32(
            /*neg_a=*/false, a, /*neg_b=*/false, b,
            /*c_mod=*/(short)0, c, /*reuse_a=*/false, /*reuse_b=*/false);

#pragma unroll
        for (int v = 0; v < 8; ++v) {
            float d2 = fmaxf(d[v], 0.0f);
            int   qg = qgv[v];
            bool valid = (qg < NQ) && (rg < NR) && (qbr[v] == rb_) && (d2 <= r2);
            if (valid) {
                float keyf = sbd ? d2 : (float)rg;   // both >= 0 -> bits monotonic
                unsigned long long pk =
                    ((unsigned long long)__float_as_uint(keyf) << 32) |
                    (unsigned int)rg;
                int pos = atomicAdd(&count[qg], 1);
                if (pos < CAP)
                    cand[(size_t)qg * CAP + pos] = pk;
            }
        }
    }
}

// ---------------------------------------------------------------------------
// Phase 2: one wave per query. Rank-select up to K smallest packed keys
// (d2 with ref-index tiebreak == JAX top_k order), emit edge rows with -1 pad.
// ---------------------------------------------------------------------------
__global__ __launch_bounds__(256)
void rg_phase2(const int* __restrict__ count,
               const unsigned long long* __restrict__ cand,
               int CAP, int NQ, int K, int* __restrict__ out, long E)
{
    __shared__ unsigned long long keys[8][CAP_MAX];
    __shared__ int oidx[8][CAP_MAX];

    const int tid  = threadIdx.x;
    const int lane = tid & 31;
    const int w    = tid >> 5;
    const int q    = blockIdx.x * 8 + w;
    const bool qa  = (q < NQ);

    int c = 0;
    if (qa) {
        c = count[q];
        if (c > CAP)     c = CAP;
        if (c > CAP_MAX) c = CAP_MAX;
    }

    unsigned long long myk[2];
#pragma unroll
    for (int t = 0; t < 2; ++t) {
        int i = lane + 32 * t;
        unsigned long long kk = ~0ull;
        if (qa && i < c) kk = cand[(size_t)q * CAP + i];
        myk[t] = kk;
        keys[w][i] = kk;
        oidx[w][i] = -1;
    }
    __syncthreads();

#pragma unroll
    for (int t = 0; t < 2; ++t) {
        if (myk[t] != ~0ull) {
            int rank = 0;
            for (int j = 0; j < CAP_MAX; ++j)
                rank += (keys[w][j] < myk[t]) ? 1 : 0;
            if (rank < K && rank < CAP_MAX)
                oidx[w][rank] = (int)(unsigned int)(myk[t] & 0xffffffffu);
        }
    }
    __syncthreads();

    if (qa) {
        for (int k = lane; k < K; k += 32) {
            int idx = (k < CAP_MAX) ? oidx[w][k] : -1;
            long base = (long)q * K + k;
            out[base]     = idx;                     // edges[0]: ref indices
            out[E + base] = (idx >= 0) ? q : -1;     // edges[1]: query indices
        }
    }
}

// ---------------------------------------------------------------------------
extern "C" void kernel_launch(void* const* d_in, const int* in_sizes, int n_in,
                              void* d_out, int out_size, void* d_ws, size_t ws_size,
                              hipStream_t stream) {
    const float* ref    = (const float*)d_in[0];   // [NR,4] (b,x,y,z)
    const float* query  = (const float*)d_in[1];   // [NQ,4]
    const int*   radius = (const int*)d_in[2];     // scalar
    // d_in[3] = num_neighbors (derived from out_size instead)
    const int*   sflag  = (const int*)d_in[4];     // scalar

    const int NR = in_sizes[0] / 4;
    const int NQ = in_sizes[1] / 4;
    if (NQ <= 0 || NR <= 0 || out_size <= 0) return;
    const int K = out_size / (2 * NQ);
    if (K <= 0) return;

    // Workspace: [counts NQ*int | packed candidates NQ*CAP*u64]
    int* count = (int*)d_ws;
    size_t cnt_bytes = (((size_t)NQ * sizeof(int)) + 255) & ~(size_t)255;
    unsigned long long* cand =
        (unsigned long long*)((char*)d_ws + cnt_bytes);

    int CAP = CAP_MAX;
    size_t avail = (ws_size > cnt_bytes) ? (ws_size - cnt_bytes) : 0;
    size_t maxcap = avail / ((size_t)NQ * sizeof(unsigned long long));
    if ((size_t)CAP > maxcap) CAP = (int)maxcap;
    if (CAP < 1) CAP = 1;

    hipMemsetAsync(count, 0, (size_t)NQ * sizeof(int), stream);

    dim3 blk(256);
    dim3 g1((NQ + 15) / 16);
    rg_phase1<<<g1, blk, 0, stream>>>(ref, query, radius, sflag,
                                      NQ, NR, CAP, count, cand);

    dim3 g2((NQ + 7) / 8);
    rg_phase2<<<g2, blk, 0, stream>>>(count, cand, CAP, NQ, K,
                                      (int*)d_out, (long)(out_size / 2));
}